// SHAN_32547262169525
// MI455X (gfx1250) — compile-verified
//
#include <hip/hip_runtime.h>

typedef __attribute__((ext_vector_type(2))) float v2f;
typedef __attribute__((ext_vector_type(8))) float v8f;

#define DIM   128
#define LLEN  200
#define SLEN  50
#define NEGV  (-1e9f)

// ---- block-wide reductions over 256 threads via LDS ----
__device__ __forceinline__ float block_max256(float v, float* red) {
  int t = threadIdx.x;
  red[t] = v; __syncthreads();
  #pragma unroll
  for (int s = 128; s > 0; s >>= 1) {
    if (t < s) red[t] = fmaxf(red[t], red[t + s]);
    __syncthreads();
  }
  float r = red[0]; __syncthreads();
  return r;
}

__device__ __forceinline__ float block_sum256(float v, float* red) {
  int t = threadIdx.x;
  red[t] = v; __syncthreads();
  #pragma unroll
  for (int s = 128; s > 0; s >>= 1) {
    if (t < s) red[t] = red[t] + red[t + s];
    __syncthreads();
  }
  float r = red[0]; __syncthreads();
  return r;
}

__global__ __launch_bounds__(256) void shan_fused_kernel(
    const float* __restrict__ user_table,   // [NUM_USERS, 128]
    const float* __restrict__ item_table,   // [NUM_ITEMS, 128]
    const float* __restrict__ W1,           // [256, 16]
    const float* __restrict__ b1,           // [16]
    const float* __restrict__ W2,           // [16, 1]
    const float* __restrict__ b2,           // [1]
    const int*   __restrict__ user_inputs,  // [B]
    const int*   __restrict__ L_inputs,     // [B, 200]
    const int*   __restrict__ S_inputs,     // [B, 50]
    const int*   __restrict__ item_inputs,  // [B]
    float*       __restrict__ out)          // [B]
{
  __shared__ float lds_seq[64 * DIM];   // stage-2 sequence (u_long + S, padded)
  __shared__ float lds_logits[208];
  __shared__ float lds_w[208];
  __shared__ float lds_su[16];
  __shared__ float lds_red[256];
  __shared__ float lds_hyb[DIM];

  const int b    = blockIdx.x;
  const int t    = threadIdx.x;
  const int lane = t & 31;
  const int wave = t >> 5;
  const int half = (lane >> 4) & 1;   // K-pair select per ISA A/B layout
  const int ln   = lane & 15;         // M (A) / N (B,D) index

  const float w2v = W2[ln];
  const float b2v = b2[0];

  // ---- preload B = W1[128:256, :] in V_WMMA_F32_16X16X4_F32 B layout ----
  // VGPR0: lanes0-15 K=0, lanes16-31 K=2 ; VGPR1: K=1 / K=3
  v2f Breg[32];
  #pragma unroll
  for (int kk = 0; kk < 32; ++kk) {
    int k0 = DIM + kk * 4 + half * 2;
    Breg[kk].x = W1[(k0 + 0) * 16 + ln];
    Breg[kk].y = W1[(k0 + 1) * 16 + ln];
  }

  // ---- su[n] = u_emb @ W1[:128, n] + b1[n]  (folded into WMMA C-init) ----
  if (t < 16) {
    long ui = (long)user_inputs[b];
    const float* urow = user_table + ui * DIM;
    float acc = b1[t];
    for (int k = 0; k < DIM; ++k) acc += urow[k] * W1[k * 16 + t];
    lds_su[t] = acc;
  }
  __syncthreads();
  const float suv = lds_su[ln];

  // ======== stage 1: long-term attention logits via f32 WMMA ========
  const int* Lrow = L_inputs + (long)b * LLEN;
  for (int tile = wave; tile < 13; tile += 8) {   // 13*16 = 208 >= 200 rows
    int m  = tile * 16 + ln;
    int mm = m < LLEN ? m : (LLEN - 1);
    int li = Lrow[mm]; li = li < 0 ? 0 : li;
    const float* arow = item_table + (long)li * DIM;

    v8f C;
    #pragma unroll
    for (int r = 0; r < 8; ++r) C[r] = suv;

    #pragma unroll
    for (int kk = 0; kk < 32; ++kk) {
      v2f A;
      A.x = arow[kk * 4 + half * 2 + 0];
      A.y = arow[kk * 4 + half * 2 + 1];
      C = __builtin_amdgcn_wmma_f32_16x16x4_f32(
              false, A, false, Breg[kk], (short)0, C, false, false);
    }

    // logit[M] = sum_n relu(C[M][n]) * W2[n] + b2 ; D layout: M = r + 8*half
    #pragma unroll
    for (int r = 0; r < 8; ++r) {
      float h = C[r] > 0.f ? C[r] : 0.f;
      float v = h * w2v;
      v += __shfl_xor(v, 1, 32);
      v += __shfl_xor(v, 2, 32);
      v += __shfl_xor(v, 4, 32);
      v += __shfl_xor(v, 8, 32);
      if (ln == 0) lds_logits[tile * 16 + half * 8 + r] = v + b2v;
    }
  }
  __syncthreads();

  // ---- masked softmax over L ----
  float lg = NEGV;
  bool  lvalid = false;
  if (t < LLEN) { lvalid = (Lrow[t] >= 0); lg = lvalid ? lds_logits[t] : NEGV; }
  float mx = block_max256(lg, lds_red);
  float e  = (t < LLEN && lvalid) ? expf(lg - mx) : 0.f;
  float sm = block_sum256(e, lds_red);
  if (t < LLEN) lds_w[t] = e / sm;
  __syncthreads();

  // ---- u_long[d] = sum_l w[l] * L_emb[l][d] (gathers hit L2-resident table) ----
  {
    int d    = t & (DIM - 1);
    int part = t >> 7;
    float acc = 0.f;
    for (int l = part * 100; l < part * 100 + 100; ++l) {
      int li = Lrow[l]; li = li < 0 ? 0 : li;
      acc += lds_w[l] * item_table[(long)li * DIM + d];
    }
    lds_red[t] = acc;
  }
  __syncthreads();

  // seq row 0 = u_long ; rows 1..50 = S_emb ; rows 51..63 = 0
  const int* Srow = S_inputs + (long)b * SLEN;
  if (t < DIM) lds_seq[t] = lds_red[t] + lds_red[t + 128];
  for (int i = t; i < 63 * DIM; i += 256) {
    int row = 1 + (i >> 7);
    int d   = i & (DIM - 1);
    float v = 0.f;
    if (row <= SLEN) {
      int si = Srow[row - 1]; si = si < 0 ? 0 : si;
      v = item_table[(long)si * DIM + d];
    }
    lds_seq[row * DIM + d] = v;
  }
  __syncthreads();

  // ======== stage 2: short-term attention logits via f32 WMMA (A from LDS) ========
  if (wave < 4) {                       // 4 tiles * 16 = 64 rows
    const float* arow = &lds_seq[(wave * 16 + ln) * DIM];
    v8f C;
    #pragma unroll
    for (int r = 0; r < 8; ++r) C[r] = suv;
    #pragma unroll
    for (int kk = 0; kk < 32; ++kk) {
      v2f A;
      A.x = arow[kk * 4 + half * 2 + 0];
      A.y = arow[kk * 4 + half * 2 + 1];
      C = __builtin_amdgcn_wmma_f32_16x16x4_f32(
              false, A, false, Breg[kk], (short)0, C, false, false);
    }
    #pragma unroll
    for (int r = 0; r < 8; ++r) {
      float h = C[r] > 0.f ? C[r] : 0.f;
      float v = h * w2v;
      v += __shfl_xor(v, 1, 32);
      v += __shfl_xor(v, 2, 32);
      v += __shfl_xor(v, 4, 32);
      v += __shfl_xor(v, 8, 32);
      if (ln == 0) lds_logits[wave * 16 + half * 8 + r] = v + b2v;
    }
  }
  __syncthreads();

  // ---- masked softmax over [u_long] + S (51 valid-ish of 64) ----
  float lg2 = NEGV;
  bool  svalid = false;
  if (t < 64) {
    if (t == 0)          svalid = true;
    else if (t <= SLEN)  svalid = (Srow[t - 1] >= 0);
    lg2 = svalid ? lds_logits[t] : NEGV;
  }
  float mx2 = block_max256(lg2, lds_red);
  float e2  = svalid ? expf(lg2 - mx2) : 0.f;
  float sm2 = block_sum256(e2, lds_red);
  if (t < 64) lds_w[t] = e2 / sm2;
  __syncthreads();

  // ---- user_hybrid[d] = sum_s w2[s] * seq[s][d] ----
  if (t < DIM) {
    float acc = 0.f;
    for (int s = 0; s <= SLEN; ++s) acc += lds_w[s] * lds_seq[s * DIM + t];
    lds_hyb[t] = acc;
  }
  __syncthreads();

  // ---- score = dot(user_hybrid, item_emb) ----
  long it = (long)item_inputs[b];
  float p = (t < DIM) ? lds_hyb[t] * item_table[it * DIM + t] : 0.f;
  float sc = block_sum256(p, lds_red);
  if (t == 0) out[b] = sc;
}

extern "C" void kernel_launch(void* const* d_in, const int* in_sizes, int n_in,
                              void* d_out, int out_size, void* d_ws, size_t ws_size,
                              hipStream_t stream) {
  const float* user_table  = (const float*)d_in[0];
  const float* item_table  = (const float*)d_in[1];
  const float* W1          = (const float*)d_in[2];
  const float* b1          = (const float*)d_in[3];
  const float* W2          = (const float*)d_in[4];
  const float* b2          = (const float*)d_in[5];
  const int*   user_inputs = (const int*)d_in[6];
  const int*   L_inputs    = (const int*)d_in[7];
  const int*   S_inputs    = (const int*)d_in[8];
  const int*   item_inputs = (const int*)d_in[9];
  float* out = (float*)d_out;

  const int B = in_sizes[6];   // 4096
  shan_fused_kernel<<<B, 256, 0, stream>>>(
      user_table, item_table, W1, b1, W2, b2,
      user_inputs, L_inputs, S_inputs, item_inputs, out);
}